// WaveDecoder_41343355191626
// MI455X (gfx1250) — compile-verified
//
#include <hip/hip_runtime.h>
#include <math.h>

// ---- problem constants (match reference) ----
#define BATCH   512
#define FEAT    8
#define IN_DIM  9          // SD + FEAT
#define R       32
#define NLAYERS 24
#define NLL     8          // layers per block (dilation cycle)
#define QL      256
#define HID     128
#define MTILE   16         // batch rows per workgroup

typedef float v2f __attribute__((ext_vector_type(2)));
typedef float v4f __attribute__((ext_vector_type(4)));
typedef float v8f __attribute__((ext_vector_type(8)));

__device__ __forceinline__ float sigm(float v) { return 1.0f / (1.0f + __expf(-v)); }

// ---------------------------------------------------------------------------
// Kernel 1: the whole recurrent network for a 16-row batch tile.
// 8 waves; wave w computes output columns [16w, 16w+16) with f32 WMMA.
// ---------------------------------------------------------------------------
__global__ __launch_bounds__(256) void wavedec_compute(
    const float* __restrict__ x,       const float* __restrict__ feat,
    const float* __restrict__ queues,
    const float* __restrict__ fc_h_w,  const float* __restrict__ fc_h_b,
    const float* __restrict__ fc_c_w,  const float* __restrict__ fc_c_b,
    const float* __restrict__ conv_w,  const float* __restrict__ conv_b,
    const float* __restrict__ fc1_w,   const float* __restrict__ fc1_b,
    const float* __restrict__ fc2_w,   const float* __restrict__ fc2_b,
    float* __restrict__ Hws,           float* __restrict__ hhat)
{
    // h_hist[0] = h0, h_hist[l+1] = layer-l output (skip). 51200 B
    __shared__ float h_hist[NLAYERS + 1][MTILE][R];
    __shared__ float past_lds[MTILE][R];        // 2 KB
    __shared__ float out_lds[MTILE][4 * R];     // 8 KB  (gate tiles / fc1 out)
    __shared__ float c_lds[MTILE][R];           // 2 KB

    const int tid  = threadIdx.x;
    const int lane = tid & 31;
    const int wave = tid >> 5;        // 0..7 -> N-tile
    const int lm   = lane & 15;
    const int hi   = lane >> 4;       // 0/1: lane half
    const int m0   = blockIdx.x * MTILE;
    const int n0   = wave * 16;

    // ---- h0 / c0: tiny 9->32 dense + tanh, scalar (2 elements / thread) ----
    for (int e = tid; e < MTILE * R; e += 256) {
        const int m = e >> 5, r = e & 31;
        const int bb = m0 + m;
        float inp[IN_DIM];
        inp[0] = x[bb];
#pragma unroll
        for (int j = 0; j < FEAT; ++j) inp[1 + j] = feat[bb * FEAT + j];
        float ah = fc_h_b[r], ac = fc_c_b[r];
#pragma unroll
        for (int j = 0; j < IN_DIM; ++j) {
            ah += inp[j] * fc_h_w[r * IN_DIM + j];
            ac += inp[j] * fc_c_w[r * IN_DIM + j];
        }
        const float h0v = tanhf(ah);
        h_hist[0][m][r] = h0v;
        c_lds[m][r]     = tanhf(ac);
        Hws[(size_t)bb * R + r] = h0v;            // H_ws slot 0
    }
    __syncthreads();

    // ---- 24-layer recurrent scan ----
    for (int l = 0; l < NLAYERS; ++l) {
        const int dil = 1 << (l & (NLL - 1));
        // gather dilated past column into LDS (strided, unavoidable)
        const float* qb = queues + (size_t)l * BATCH * R * QL + (QL - dil);
        for (int e = tid; e < MTILE * R; e += 256) {
            const int m = e >> 5, r = e & 31;
            past_lds[m][r] = qb[((size_t)(m0 + m) * R + r) * QL];
        }
        __syncthreads();

        // out[m][n] = past·W0^T + h·W1^T  via V_WMMA_F32_16X16X4_F32, K=32
        const float* Wl = conv_w + (size_t)l * (4 * R) * R * 2;   // [128][32][2]
        v8f acc = {};
#pragma unroll
        for (int k0 = 0; k0 < R; k0 += 4) {
            const int ka = k0 + 2 * hi;
            v2f a0 = { past_lds[lm][ka], past_lds[lm][ka + 1] };
            v2f b0 = { Wl[((n0 + lm) * R + ka) * 2 + 0],
                       Wl[((n0 + lm) * R + ka + 1) * 2 + 0] };
            acc = __builtin_amdgcn_wmma_f32_16x16x4_f32(false, a0, false, b0,
                                                        (short)0, acc, false, false);
            v2f a1 = { h_hist[l][lm][ka], h_hist[l][lm][ka + 1] };
            v2f b1 = { Wl[((n0 + lm) * R + ka) * 2 + 1],
                       Wl[((n0 + lm) * R + ka + 1) * 2 + 1] };
            acc = __builtin_amdgcn_wmma_f32_16x16x4_f32(false, a1, false, b1,
                                                        (short)0, acc, false, false);
        }
        const float bias = conv_b[l * 4 * R + n0 + lm];
#pragma unroll
        for (int v = 0; v < 8; ++v)
            out_lds[v + 8 * hi][n0 + lm] = acc[v] + bias;
        __syncthreads();

        // LSTM-style gates (ig | cf | cg | eg), 2 elements / thread
        for (int e = tid; e < MTILE * R; e += 256) {
            const int m = e >> 5, r = e & 31;
            const float ig = out_lds[m][r];
            const float cf = out_lds[m][r + R];
            const float cg = out_lds[m][r + 2 * R];
            const float eg = out_lds[m][r + 3 * R];
            const float cn = sigm(ig) * c_lds[m][r] + tanhf(cf) * sigm(cg);
            const float hn = sigm(eg) * tanhf(cn);
            c_lds[m][r] = cn;
            h_hist[l + 1][m][r] = hn;
            Hws[(size_t)(l + 1) * BATCH * R + (size_t)(m0 + m) * R + r] = hn;
        }
        __syncthreads();
    }

    // ---- fc1 (768 -> 128) with WMMA over the LDS skip history, + ReLU ----
    {
        v8f acc = {};
        for (int kt = 0; kt < NLAYERS; ++kt) {
#pragma unroll
            for (int k0 = 0; k0 < R; k0 += 4) {
                const int ka = k0 + 2 * hi;
                const int kk = kt * R + ka;
                v2f a = { h_hist[1 + kt][lm][ka], h_hist[1 + kt][lm][ka + 1] };
                v2f b = { fc1_w[(n0 + lm) * (NLAYERS * R) + kk],
                          fc1_w[(n0 + lm) * (NLAYERS * R) + kk + 1] };
                acc = __builtin_amdgcn_wmma_f32_16x16x4_f32(false, a, false, b,
                                                            (short)0, acc, false, false);
            }
        }
        const float bias = fc1_b[n0 + lm];
#pragma unroll
        for (int v = 0; v < 8; ++v) {
            const float yv = acc[v] + bias;
            out_lds[v + 8 * hi][n0 + lm] = yv > 0.0f ? yv : 0.0f;
        }
    }
    __syncthreads();

    // ---- fc2 (128 -> 1): 16 tiny dot products ----
    if (tid < MTILE) {
        float s = fc2_b[0];
#pragma unroll 8
        for (int n = 0; n < HID; ++n) s += out_lds[tid][n] * fc2_w[n];
        hhat[m0 + tid] = s;
    }
}

// ---------------------------------------------------------------------------
// Kernel 2: update_queues = concat(queues, h_seq) along t. Pure streaming
// copy (~800 MB traffic) — this IS the roofline. 4 rows / 256-thread block,
// 16B coalesced loads. Both sides are touched exactly once, so use
// non-temporal hints (ISA §7.3 TH=NT) to keep the 600+ MB single-use stream
// from displacing resident data in the 192 MB L2.
// ---------------------------------------------------------------------------
__global__ __launch_bounds__(256) void wavedec_queues(
    const float* __restrict__ queues, const float* __restrict__ Hws,
    float* __restrict__ out)   // out = d_out; update_queues begins at +BATCH
{
    const int tid = threadIdx.x;
    const int sub = tid >> 6;          // row within block (0..3)
    const int l64 = tid & 63;          // 64 lanes x 4 floats = 256 floats
    const size_t row = (size_t)blockIdx.x * 4 + sub;   // < 24*512*32

    const v4f v =
        __builtin_nontemporal_load((const v4f*)(queues + row * QL + l64 * 4));
    float* dst = out + BATCH + row * (QL + 1) + l64 * 4;
    __builtin_nontemporal_store(v.x, dst + 0);
    __builtin_nontemporal_store(v.y, dst + 1);
    __builtin_nontemporal_store(v.z, dst + 2);
    __builtin_nontemporal_store(v.w, dst + 3);
    if (l64 == 0)
        __builtin_nontemporal_store(Hws[row],
                                    out + BATCH + row * (QL + 1) + QL);
}

extern "C" void kernel_launch(void* const* d_in, const int* in_sizes, int n_in,
                              void* d_out, int out_size, void* d_ws, size_t ws_size,
                              hipStream_t stream)
{
    const float* x      = (const float*)d_in[0];
    const float* feat   = (const float*)d_in[1];
    const float* queues = (const float*)d_in[2];
    const float* fc_h_w = (const float*)d_in[3];
    const float* fc_h_b = (const float*)d_in[4];
    const float* fc_c_w = (const float*)d_in[5];
    const float* fc_c_b = (const float*)d_in[6];
    const float* conv_w = (const float*)d_in[7];
    const float* conv_b = (const float*)d_in[8];
    const float* fc1_w  = (const float*)d_in[9];
    const float* fc1_b  = (const float*)d_in[10];
    const float* fc2_w  = (const float*)d_in[11];
    const float* fc2_b  = (const float*)d_in[12];

    float* out = (float*)d_out;
    float* Hws = (float*)d_ws;         // (NLAYERS+1) * BATCH * R floats = 1.6 MB

    wavedec_compute<<<BATCH / MTILE, 256, 0, stream>>>(
        x, feat, queues, fc_h_w, fc_h_b, fc_c_w, fc_c_b,
        conv_w, conv_b, fc1_w, fc1_b, fc2_w, fc2_b, Hws, out);

    const int nrows = NLAYERS * BATCH * R;        // 393216
    wavedec_queues<<<nrows / 4, 256, 0, stream>>>(queues, Hws, out);
}